// SingleLIFLayer_4664334483974
// MI455X (gfx1250) — compile-verified
//
#include <hip/hip_runtime.h>

// ---------------------------------------------------------------------------
// LIF recurrence, MI455X (gfx1250): memory-bound streaming kernel.
//   V <- V + alpha*(I - V);  spike = V >= 1;  V <- 0 where spiked
// T=1024 sequential steps, N=65536 independent neurons.
//
// Roofline: 256 MB in + 256 MB out @ 23.3 TB/s => ~22 us floor; ~0.27 GFLOP
// total => pure streaming problem. Strategy: per-wave 48-deep async
// GLOBAL->LDS pipeline (ASYNCcnt) => 48 * 256 KB = 12 MB of HBM traffic in
// flight (covers ~500 ns latency at 23.3 TB/s); ds_load_b128 consumption;
// non-temporal b128 stores for the write-once 256 MB output (don't thrash
// the 192 MB L2). 128 blocks x 128 threads spreads VALU work across WGPs so
// instruction issue stays well under the memory floor.
// ---------------------------------------------------------------------------

#define N_NEURONS 65536
#define T_STEPS   1024
#define BLOCK     128
#define VPT       4            // neurons (floats) per thread -> b128 accesses
#define DEPTH     48           // async pipeline depth (< 63: ASYNCcnt is 6-bit)

typedef float floatx4 __attribute__((ext_vector_type(4)));

#if defined(__AMDGCN__)

// Builtin signature (probe-confirmed via hipcc diagnostic): params are
// v4i (gcc vector_size(16) int) pointers in AS1 (global) and AS3 (LDS),
// then immediate offset and immediate cpol.
typedef int v4i __attribute__((vector_size(4 * sizeof(int))));
typedef __attribute__((address_space(1))) v4i gv4i_t;
typedef __attribute__((address_space(3))) v4i lv4i_t;

#if __has_builtin(__builtin_amdgcn_global_load_async_to_lds_b128)
#define ASYNC_B128(gp, lp)                                                     \
  __builtin_amdgcn_global_load_async_to_lds_b128((gv4i_t*)(gp),                \
                                                 (lv4i_t*)(lp), 0, 0)
#else
#define ASYNC_B128(gp, lp)                                                     \
  do {                                                                         \
    unsigned __lds_off =                                                       \
        (unsigned)(unsigned long long)(__attribute__((address_space(3)))       \
                                       void*)(lp);                             \
    unsigned long long __gaddr = (unsigned long long)(gp);                     \
    asm volatile("global_load_async_to_lds_b128 %0, %1, off"                   \
                 :: "v"(__lds_off), "v"(__gaddr)                               \
                 : "memory");                                                  \
  } while (0)
#endif

#if __has_builtin(__builtin_amdgcn_s_wait_asynccnt)
#define WAIT_ASYNC(n) __builtin_amdgcn_s_wait_asynccnt(n)
#else
#define WAIT_ASYNC(n) asm volatile("s_wait_asynccnt %0" :: "i"(n) : "memory")
#endif

#if __has_builtin(__builtin_amdgcn_s_wait_dscnt)
#define WAIT_DS(n) __builtin_amdgcn_s_wait_dscnt(n)
#else
#define WAIT_DS(n) asm volatile("s_wait_dscnt %0" :: "i"(n) : "memory")
#endif

#else  // host pass: bodies are parsed but never codegen'd
#define ASYNC_B128(gp, lp) ((void)0)
#define WAIT_ASYNC(n)      ((void)0)
#define WAIT_DS(n)         ((void)0)
#endif

__device__ __forceinline__ floatx4 lif_step(floatx4& V, floatx4 cur) {
  const float ALPHA   = 0.05f;   // DT / TAU = 1/20
  const float V_TH    = 1.0f;
  const float V_RESET = 0.0f;
  floatx4 sp;
#pragma unroll
  for (int k = 0; k < VPT; ++k) {
    float v = V[k];
    v = __builtin_fmaf(ALPHA, cur[k] - v, v);   // V += alpha*(I - V)
    const bool spike = (v >= V_TH);
    sp[k] = spike ? 1.0f : 0.0f;
    V[k]  = spike ? V_RESET : v;
  }
  return sp;
}

__global__ __launch_bounds__(BLOCK) void
lif_scan_kernel(const float* __restrict__ in, float* __restrict__ out) {
  __shared__ floatx4 ring[DEPTH * BLOCK];   // 48 stages x 2 KB = 96 KB

  const int tid  = threadIdx.x;
  const int gidx = blockIdx.x * BLOCK + tid;          // float4 slot index
  const float* src = in  + (size_t)gidx * VPT;        // column base, row 0
  float*       dst = out + (size_t)gidx * VPT;

  // ---- prologue: fill the async pipeline -------------------------------
#pragma unroll
  for (int s = 0; s < DEPTH; ++s) {
    ASYNC_B128(src + (size_t)s * N_NEURONS, &ring[s * BLOCK + tid]);
  }

  floatx4 V = {0.0f, 0.0f, 0.0f, 0.0f};               // V_RESET
  int slot = 0;                                       // uniform ring cursor

  // ---- steady state: wait oldest stage, consume, refill ----------------
  int t = 0;
  for (; t < T_STEPS - DEPTH; ++t) {
    WAIT_ASYNC(DEPTH - 1);                 // oldest async load has landed
    floatx4 cur = ring[slot * BLOCK + tid];           // ds_load_b128
    floatx4 sp  = lif_step(V, cur);
    __builtin_nontemporal_store(sp, (floatx4*)(dst + (size_t)t * N_NEURONS));
    WAIT_DS(0);                            // slot fully read before reuse
    ASYNC_B128(src + (size_t)(t + DEPTH) * N_NEURONS,
               &ring[slot * BLOCK + tid]);
    slot = (slot + 1 == DEPTH) ? 0 : slot + 1;
  }

  // ---- epilogue: drain remaining DEPTH stages --------------------------
  WAIT_ASYNC(0);
  for (; t < T_STEPS; ++t) {
    floatx4 cur = ring[slot * BLOCK + tid];
    floatx4 sp  = lif_step(V, cur);
    __builtin_nontemporal_store(sp, (floatx4*)(dst + (size_t)t * N_NEURONS));
    slot = (slot + 1 == DEPTH) ? 0 : slot + 1;
  }
}

extern "C" void kernel_launch(void* const* d_in, const int* in_sizes, int n_in,
                              void* d_out, int out_size, void* d_ws, size_t ws_size,
                              hipStream_t stream) {
  (void)in_sizes; (void)n_in; (void)out_size; (void)d_ws; (void)ws_size;
  const float* input = (const float*)d_in[0];   // (T, N) float32
  float* spikes      = (float*)d_out;           // (T, N) float32

  const int blocks = N_NEURONS / (BLOCK * VPT); // 128 blocks x 128 threads
  lif_scan_kernel<<<blocks, BLOCK, 0, stream>>>(input, spikes);
}